// CubicalModel_ISM_29214367547808
// MI455X (gfx1250) — compile-verified
//
#include <hip/hip_runtime.h>

typedef __attribute__((ext_vector_type(2))) float v2f;
typedef __attribute__((ext_vector_type(8))) float v8f;

#define KDIM   32768
#define NWAVES 32
#define KSLICE (KDIM / NWAVES)   // 1024 k-values per wave

// One block = one tile of 16 dot products (16 gathered rows of I vs p).
// blockIdx.x in [0,14): blocks 0..6 -> image 1, blocks 7..13 -> image 2.
__global__ __launch_bounds__(NWAVES * 32)
void diagram_dot_wmma(const float* __restrict__ p0, const float* __restrict__ p1,
                      const float* __restrict__ I1, const float* __restrict__ I2,
                      const int* __restrict__ inds1, const int* __restrict__ inds2,
                      float* __restrict__ out)
{
    const int blk      = blockIdx.x;      // 0..13
    const int img      = blk / 7;         // 0 or 1
    const int tile     = blk % 7;         // 0..6
    const int tileBase = tile * 16;       // output index base within this image

    const float* __restrict__ I    = (img == 0) ? I1 : I2;
    const float* __restrict__ p    = (img == 0) ? p0 : p1;
    const int*   __restrict__ inds = (img == 0) ? inds1 : inds2;

    const int tid  = (int)threadIdx.x;
    const int lane = tid & 31;
    const int wave = tid >> 5;
    const int m    = lane & 15;           // matrix row this lane carries for A
    const int hb   = lane >> 4;           // half of wave: 0 -> K=0,1 ; 1 -> K=2,3
    const int koff = hb * 2;

    // Gather-first: resolve which row of I this lane's output needs.
    int j = tileBase + m;                 // output index within image (0..111)
    int e = 0;                            // padded lanes compute a dummy row 0
    if (j < 100) e = inds[2 * j] * 28 + inds[2 * j + 1];
    const float* rowPtr = I + (size_t)e * (size_t)KDIM;

    const int kstart = wave * KSLICE;

    v8f acc = {0.f, 0.f, 0.f, 0.f, 0.f, 0.f, 0.f, 0.f};

    // A(16x4): lane L<16 holds row m, k={kc,kc+1}; lane L>=16 holds row m, k={kc+2,kc+3}.
    // B(4x16): B[k,n] = p[kc+k] for every column n -> same per-lane pair pattern as A,
    //          sourced from p. All 16 columns of D then hold the dot product.
    #pragma unroll 4
    for (int kc = 0; kc < KSLICE; kc += 4) {
        const int k = kstart + kc + koff;
        v2f a = *(const v2f*)(rowPtr + k);
        v2f b = *(const v2f*)(p + k);
        acc = __builtin_amdgcn_wmma_f32_16x16x4_f32(
            /*neg_a=*/false, a, /*neg_b=*/false, b,
            /*c_mod=*/(short)0, acc, /*reuse_a=*/false, /*reuse_b=*/false);
    }

    // Cross-wave reduction: lane 0 holds D[r,0] = rows 0..7, lane 16 holds rows 8..15.
    __shared__ float red[NWAVES * 16];
    if (lane == 0) {
        #pragma unroll
        for (int r = 0; r < 8; ++r) red[wave * 16 + r] = acc[r];
    } else if (lane == 16) {
        #pragma unroll
        for (int r = 0; r < 8; ++r) red[wave * 16 + 8 + r] = acc[r];
    }
    __syncthreads();

    if (tid < 16) {
        float s = 0.f;
        #pragma unroll
        for (int w = 0; w < NWAVES; ++w) s += red[w * 16 + tid];
        const int jj = tileBase + tid;
        if (jj < 100) out[img * 100 + jj] = s;
    }
}

extern "C" void kernel_launch(void* const* d_in, const int* in_sizes, int n_in,
                              void* d_out, int out_size, void* d_ws, size_t ws_size,
                              hipStream_t stream) {
    (void)in_sizes; (void)n_in; (void)out_size; (void)d_ws; (void)ws_size;
    const float* p0    = (const float*)d_in[0];
    const float* p1    = (const float*)d_in[1];
    const float* I1    = (const float*)d_in[2];
    const float* I2    = (const float*)d_in[3];
    const int*   inds1 = (const int*)d_in[4];
    const int*   inds2 = (const int*)d_in[5];
    float* out = (float*)d_out;

    dim3 grid(14);                 // 2 images x 7 tiles of 16 outputs (covers 100 each)
    dim3 block(NWAVES * 32);       // 32 wave32 waves splitting K = 32768
    diagram_dot_wmma<<<grid, block, 0, stream>>>(p0, p1, I1, I2, inds1, inds2, out);
}